// GraphSAGERecommender_85031762526359
// MI455X (gfx1250) — compile-verified
//
#include <hip/hip_runtime.h>
#include <math.h>

typedef __attribute__((ext_vector_type(2))) float v2f;
typedef __attribute__((ext_vector_type(8))) float v8f;

#define NNODES 100000

// ---------------------------------------------------------------- degree
__global__ void degree_kernel(const int* __restrict__ dst, float* __restrict__ deg, int E) {
    int t = blockIdx.x * blockDim.x + threadIdx.x;
    if (t < E) atomicAdd(&deg[dst[t]], 1.0f);
}

// ------------------------------------------------------- scatter-add (mean agg)
// One thread per (edge, 4-feature chunk). Feature tables are L2-resident.
__global__ void scatter_add_kernel(const float* __restrict__ feat,
                                   const int* __restrict__ src,
                                   const int* __restrict__ dst,
                                   float* __restrict__ agg,
                                   int E, int C4 /* features/4 */) {
    long long t = (long long)blockIdx.x * blockDim.x + threadIdx.x;
    long long total = (long long)E * C4;
    if (t >= total) return;
    int e = (int)(t / C4);
    int c = (int)(t % C4);
    int s = src[e], d = dst[e];
    float4 v = reinterpret_cast<const float4*>(feat + (long long)s * (C4 * 4))[c];
    float* p = agg + (long long)d * (C4 * 4) + (long long)c * 4;
    atomicAdd(p + 0, v.x); atomicAdd(p + 1, v.y);
    atomicAdd(p + 2, v.z); atomicAdd(p + 3, v.w);
}

// ------------------------------------------------------------- agg -> mean
__global__ void mean_div_kernel(float* __restrict__ agg, const float* __restrict__ deg,
                                long long total, int F) {
    long long t = (long long)blockIdx.x * blockDim.x + threadIdx.x;
    if (t >= total) return;
    float d = deg[t / F];
    d = d > 1.0f ? d : 1.0f;
    agg[t] = agg[t] / d;
}

// ------------------------------------------------- fused SAGE GEMM via WMMA f32
// out = act( A1 @ W1^T + A2 @ W2^T + bias ), W* are (N,K) row-major.
// One wave32 computes one 16x16 output tile using V_WMMA_F32_16X16X4_F32.
// Grid is exact (no partial tiles) so EXEC is all-1s at every WMMA.
template <int K, bool RELU>
__global__ void sage_gemm_wmma(const float* __restrict__ A1,
                               const float* __restrict__ A2,
                               const float* __restrict__ W1,
                               const float* __restrict__ W2,
                               const float* __restrict__ bias,
                               float* __restrict__ out,
                               int NT /* N/16 */, int N) {
    int wid  = (blockIdx.x * blockDim.x + threadIdx.x) >> 5;
    int lane = threadIdx.x & 31;
    int mt = wid / NT, nt = wid % NT;
    int m0 = mt << 4, n0 = nt << 4;

    int l15   = lane & 15;
    int half2 = (lane >> 4) << 1;  // 0 for lanes 0-15 (K=0,1), 2 for lanes 16-31 (K=2,3)

    const float* a1p = A1 + (long long)(m0 + l15) * K + half2;
    const float* a2p = A2 + (long long)(m0 + l15) * K + half2;
    const float* w1p = W1 + (long long)(n0 + l15) * K + half2;
    const float* w2p = W2 + (long long)(n0 + l15) * K + half2;

    v8f c = {};
#pragma unroll 8
    for (int k0 = 0; k0 < K; k0 += 4) {
        float2 af = *reinterpret_cast<const float2*>(a1p + k0);
        float2 wf = *reinterpret_cast<const float2*>(w1p + k0);
        v2f a = {af.x, af.y};
        v2f b = {wf.x, wf.y};
        c = __builtin_amdgcn_wmma_f32_16x16x4_f32(false, a, false, b, (short)0, c, false, false);
    }
#pragma unroll 8
    for (int k0 = 0; k0 < K; k0 += 4) {
        float2 af = *reinterpret_cast<const float2*>(a2p + k0);
        float2 wf = *reinterpret_cast<const float2*>(w2p + k0);
        v2f a = {af.x, af.y};
        v2f b = {wf.x, wf.y};
        c = __builtin_amdgcn_wmma_f32_16x16x4_f32(false, a, false, b, (short)0, c, false, false);
    }

    float bv = bias[n0 + l15];
    int rbase = m0 + ((lane >> 4) << 3);  // lanes 16-31 hold rows M=8..15 of the tile
#pragma unroll
    for (int v = 0; v < 8; ++v) {
        float r = c[v] + bv;
        if (RELU) r = r > 0.0f ? r : 0.0f;
        out[(long long)(rbase + v) * N + n0 + l15] = r;
    }
}

// ---------------------------------------------------- link predictor (wave/pair)
__global__ void link_pred_kernel(const float* __restrict__ h2,
                                 const int* __restrict__ pairs,
                                 const float* __restrict__ Wlp,
                                 const float* __restrict__ blp,
                                 float* __restrict__ out, int P) {
    int wave = (blockIdx.x * blockDim.x + threadIdx.x) >> 5;
    int lane = threadIdx.x & 31;
    if (wave >= P) return;
    int s = pairs[2 * wave + 0];
    int d = pairs[2 * wave + 1];
    float4 vs = reinterpret_cast<const float4*>(h2 + (long long)s * 128)[lane];
    float4 vd = reinterpret_cast<const float4*>(h2 + (long long)d * 128)[lane];
    float4 ws = reinterpret_cast<const float4*>(Wlp)[lane];
    float4 wd = reinterpret_cast<const float4*>(Wlp + 128)[lane];
    float p = vs.x * ws.x + vs.y * ws.y + vs.z * ws.z + vs.w * ws.w
            + vd.x * wd.x + vd.y * wd.y + vd.z * wd.z + vd.w * wd.w;
#pragma unroll
    for (int off = 16; off > 0; off >>= 1) p += __shfl_xor(p, off, 32);
    if (lane == 0) out[wave] = 1.0f / (1.0f + expf(-(p + blp[0])));
}

// ---------------------------------------------------------------- launcher
extern "C" void kernel_launch(void* const* d_in, const int* in_sizes, int n_in,
                              void* d_out, int out_size, void* d_ws, size_t ws_size,
                              hipStream_t stream) {
    const float* x   = (const float*)d_in[0];
    const int*   ei  = (const int*)d_in[1];
    const int* pairs = (const int*)d_in[2];
    const float* Wl1 = (const float*)d_in[3];
    const float* Wr1 = (const float*)d_in[4];
    const float* b1  = (const float*)d_in[5];
    const float* Wl2 = (const float*)d_in[6];
    const float* Wr2 = (const float*)d_in[7];
    const float* b2  = (const float*)d_in[8];
    const float* Wlp = (const float*)d_in[9];
    const float* blp = (const float*)d_in[10];

    const int E = in_sizes[1] / 2;   // 1,600,000
    const int P = in_sizes[2] / 2;   // 200,000
    const int M = NNODES;            // 100,000 = 6250 * 16

    const int* src = ei;
    const int* dst = ei + E;

    // workspace layout (256B aligned slices)
    char* w = (char*)d_ws;
    auto alloc = [&](size_t bytes) {
        char* p = w;
        w += (bytes + 255) & ~(size_t)255;
        return p;
    };
    float* deg  = (float*)alloc(sizeof(float) * (size_t)M);         //   0.4 MB
    float* agg1 = (float*)alloc(sizeof(float) * (size_t)M * 128);   //  51.2 MB (becomes mean1)
    float* h1   = (float*)alloc(sizeof(float) * (size_t)M * 256);   // 102.4 MB
    float* agg2 = (float*)alloc(sizeof(float) * (size_t)M * 256);   // 102.4 MB (becomes mean2)
    float* h2   = (float*)alloc(sizeof(float) * (size_t)M * 128);   //  51.2 MB

    hipMemsetAsync(deg,  0, sizeof(float) * (size_t)M,       stream);
    hipMemsetAsync(agg1, 0, sizeof(float) * (size_t)M * 128, stream);
    hipMemsetAsync(agg2, 0, sizeof(float) * (size_t)M * 256, stream);

    // degree (shared by both layers)
    degree_kernel<<<(E + 255) / 256, 256, 0, stream>>>(dst, deg, E);

    // ---- layer 1: mean agg of x, then h1 = relu(mean@Wl1^T + x@Wr1^T + b1)
    {
        long long total = (long long)E * 32;  // 128/4 chunks
        scatter_add_kernel<<<(int)((total + 255) / 256), 256, 0, stream>>>(x, src, dst, agg1, E, 32);
    }
    mean_div_kernel<<<(int)(((long long)M * 128 + 255) / 256), 256, 0, stream>>>(
        agg1, deg, (long long)M * 128, 128);
    // waves = (M/16)*(256/16) = 100000 ; blocks = 100000*32/256 = 12500 (exact)
    sage_gemm_wmma<128, true><<<12500, 256, 0, stream>>>(agg1, x, Wl1, Wr1, b1, h1, 16, 256);

    // ---- layer 2: mean agg of h1, then h2 = mean@Wl2^T + h1@Wr2^T + b2
    {
        long long total = (long long)E * 64;  // 256/4 chunks
        scatter_add_kernel<<<(int)((total + 255) / 256), 256, 0, stream>>>(h1, src, dst, agg2, E, 64);
    }
    mean_div_kernel<<<(int)(((long long)M * 256 + 255) / 256), 256, 0, stream>>>(
        agg2, deg, (long long)M * 256, 256);
    // waves = (M/16)*(128/16) = 50000 ; blocks = 50000*32/256 = 6250 (exact)
    sage_gemm_wmma<256, false><<<6250, 256, 0, stream>>>(agg2, h1, Wl2, Wr2, b2, h2, 8, 128);

    // ---- link prediction: one wave per pair
    link_pred_kernel<<<P * 32 / 256, 256, 0, stream>>>(h2, pairs, Wlp, blp, (float*)d_out, P);
}